// PhysicsGNN_NC_9792525434960
// MI455X (gfx1250) — compile-verified
//
#include <hip/hip_runtime.h>
#include <math.h>

#define N_NODES   50000
#define N_EDGES   1600000
#define NFEAT     500
#define HIDDEN    64
#define NCLASS    40
#define NUM_LAYERS 4
#define STEP_SZ   0.1f
#define M_TILES   (N_NODES / 16)   // 3125, exact

typedef __attribute__((ext_vector_type(2))) float v2f;
typedef __attribute__((ext_vector_type(8))) float v8f;

// ---------------- degree / coef / W_eff ----------------

__global__ void deg_count_kernel(const int* __restrict__ col, float* __restrict__ deg) {
  int e = blockIdx.x * blockDim.x + threadIdx.x;
  if (e < N_EDGES) atomicAdd(&deg[col[e]], 1.0f);
}

__global__ void deg_inv_kernel(float* __restrict__ deg) {
  int i = blockIdx.x * blockDim.x + threadIdx.x;
  if (i < N_NODES) {
    float d = deg[i];
    deg[i] = (d > 0.0f) ? rsqrtf(d) : 0.0f;
  }
}

__global__ void coef_kernel(const int* __restrict__ row, const int* __restrict__ col,
                            const float* __restrict__ dinv, float* __restrict__ coef) {
  int e = blockIdx.x * blockDim.x + threadIdx.x;
  if (e < N_EDGES) coef[e] = dinv[row[e]] * dinv[col[e]];
}

// W_eff = sym(triu(pw[:, :-2],1)) + diag(q * rowsum|W0| + r)
__global__ void weff_kernel(const float* __restrict__ pw, float* __restrict__ W) {
  int i = threadIdx.x;                  // 64 threads
  const int LD = HIDDEN + 2;
  float q = pw[i * LD + HIDDEN];
  float r = pw[i * LD + HIDDEN + 1];
  float s = 0.0f;
  for (int j = 0; j < HIDDEN; ++j) {
    float v = 0.0f;
    if (i < j)      v = pw[i * LD + j];
    else if (i > j) v = pw[j * LD + i];
    s += fabsf(v);
    W[i * HIDDEN + j] = v;
  }
  W[i * HIDDEN + i] = q * s + r;
}

// ---------------- encoder GEMM: h = x @ enc_w.T (50000x500 @ 500x64) ----------------
// one wave per 16-node tile, full N=64 via 4 f32 WMMA accumulators, K step 4

__global__ __launch_bounds__(128)
void enc_gemm_kernel(const float* __restrict__ x, const float* __restrict__ ew,
                     float* __restrict__ h, float* __restrict__ x0) {
  int wave = (int)((blockIdx.x * blockDim.x + threadIdx.x) >> 5);
  if (wave >= M_TILES) return;          // wave-uniform exit: EXEC stays all-1
  int lane = threadIdx.x & 31;
  int mr = lane & 15;                   // M within tile (A) / N within tile (B)
  int kh = lane >> 4;                   // K-half select
  const float* arow = x  + (size_t)(wave * 16 + mr) * NFEAT;
  const float* b0r  = ew + (size_t)(mr +  0) * NFEAT;
  const float* b1r  = ew + (size_t)(mr + 16) * NFEAT;
  const float* b2r  = ew + (size_t)(mr + 32) * NFEAT;
  const float* b3r  = ew + (size_t)(mr + 48) * NFEAT;
  v8f c0 = {}, c1 = {}, c2 = {}, c3 = {};
  for (int k = 0; k < NFEAT; k += 4) {
    int ko = k + 2 * kh;
    v2f a  = *(const v2f*)(arow + ko);
    v2f b0 = *(const v2f*)(b0r + ko);
    v2f b1 = *(const v2f*)(b1r + ko);
    v2f b2 = *(const v2f*)(b2r + ko);
    v2f b3 = *(const v2f*)(b3r + ko);
    c0 = __builtin_amdgcn_wmma_f32_16x16x4_f32(false, a, false, b0, (short)0, c0, false, false);
    c1 = __builtin_amdgcn_wmma_f32_16x16x4_f32(false, a, false, b1, (short)0, c1, false, false);
    c2 = __builtin_amdgcn_wmma_f32_16x16x4_f32(false, a, false, b2, (short)0, c2, false, false);
    c3 = __builtin_amdgcn_wmma_f32_16x16x4_f32(false, a, false, b3, (short)0, c3, false, false);
  }
  int mbase = wave * 16 + kh * 8;       // C layout: vgpr r -> M = r (lanes 0-15) / 8+r (16-31)
  for (int r = 0; r < 8; ++r) {
    size_t o = (size_t)(mbase + r) * HIDDEN + mr;
    h[o]      = c0[r]; x0[o]      = c0[r];
    h[o + 16] = c1[r]; x0[o + 16] = c1[r];
    h[o + 32] = c2[r]; x0[o + 32] = c2[r];
    h[o + 48] = c3[r]; x0[o + 48] = c3[r];
  }
}

// ---------------- per-layer GEMM: out_p = h @ W_eff.T (W_eff staged in LDS) ----------------

__global__ __launch_bounds__(128)
void hw_gemm_kernel(const float* __restrict__ h, const float* __restrict__ W,
                    float* __restrict__ outp) {
  __shared__ float Ws[HIDDEN * HIDDEN]; // 16 KB of the 320 KB/WGP LDS
  for (int i = threadIdx.x; i < HIDDEN * HIDDEN; i += blockDim.x) Ws[i] = W[i];
  __syncthreads();
  int wave = (int)((blockIdx.x * blockDim.x + threadIdx.x) >> 5);
  if (wave >= M_TILES) return;
  int lane = threadIdx.x & 31;
  int mr = lane & 15;
  int kh = lane >> 4;
  const float* arow = h + (size_t)(wave * 16 + mr) * HIDDEN;
  v8f c0 = {}, c1 = {}, c2 = {}, c3 = {};
  for (int k = 0; k < HIDDEN; k += 4) {
    int ko = k + 2 * kh;
    v2f a  = *(const v2f*)(arow + ko);
    v2f b0 = *(const v2f*)(&Ws[(mr +  0) * HIDDEN + ko]);
    v2f b1 = *(const v2f*)(&Ws[(mr + 16) * HIDDEN + ko]);
    v2f b2 = *(const v2f*)(&Ws[(mr + 32) * HIDDEN + ko]);
    v2f b3 = *(const v2f*)(&Ws[(mr + 48) * HIDDEN + ko]);
    c0 = __builtin_amdgcn_wmma_f32_16x16x4_f32(false, a, false, b0, (short)0, c0, false, false);
    c1 = __builtin_amdgcn_wmma_f32_16x16x4_f32(false, a, false, b1, (short)0, c1, false, false);
    c2 = __builtin_amdgcn_wmma_f32_16x16x4_f32(false, a, false, b2, (short)0, c2, false, false);
    c3 = __builtin_amdgcn_wmma_f32_16x16x4_f32(false, a, false, b3, (short)0, c3, false, false);
  }
  int mbase = wave * 16 + kh * 8;
  for (int r = 0; r < 8; ++r) {
    size_t o = (size_t)(mbase + r) * HIDDEN + mr;
    outp[o]      = c0[r];
    outp[o + 16] = c1[r];
    outp[o + 32] = c2[r];
    outp[o + 48] = c3[r];
  }
}

// ---------------- edge aggregation: agg[col] += coef * out_p[row] ----------------
// one wave32 per edge; wave-uniform metadata scalarized via readfirstlane so the
// hot loads/atomics use SGPR base + constant offset; one b64 load per lane.

__global__ __launch_bounds__(256)
void edge_agg_kernel(const int* __restrict__ row, const int* __restrict__ col,
                     const float* __restrict__ coef, const float* __restrict__ outp,
                     float* __restrict__ agg) {
  int e = (int)((blockIdx.x * blockDim.x + threadIdx.x) >> 5);
  if (e >= N_EDGES) return;
  int lane = threadIdx.x & 31;
  int r  = __builtin_amdgcn_readfirstlane(row[e]);
  int c  = __builtin_amdgcn_readfirstlane(col[e]);
  float cf = __int_as_float(__builtin_amdgcn_readfirstlane(__float_as_int(coef[e])));
  const float* src = outp + (size_t)r * HIDDEN + 2 * lane;
  float*       dst = agg  + (size_t)c * HIDDEN + 2 * lane;
  v2f v = *(const v2f*)src;             // single global_load_b64 per lane
  atomicAdd(&dst[0], cf * v.x);
  atomicAdd(&dst[1], cf * v.y);
}

// ---------------- node update: h += STEP * relu(agg - h*ext_w + x0*beta) ----------------

__global__ __launch_bounds__(256)
void update_kernel(float* __restrict__ h, const float* __restrict__ agg,
                   const float* __restrict__ x0, const float* __restrict__ ext_w,
                   const float* __restrict__ beta) {
  int idx = blockIdx.x * blockDim.x + threadIdx.x; // exactly N_NODES*HIDDEN threads
  int j = idx & (HIDDEN - 1);
  float hv = h[idx];
  float conv = agg[idx] - hv * ext_w[j] + x0[idx] * beta[0];
  h[idx] = hv + STEP_SZ * fmaxf(conv, 0.0f);
}

// ---------------- decoder GEMM: out = h @ dec_w.T (N=40, padded to 48) ----------------

__global__ __launch_bounds__(128)
void dec_gemm_kernel(const float* __restrict__ h, const float* __restrict__ dw,
                     float* __restrict__ out) {
  int wave = (int)((blockIdx.x * blockDim.x + threadIdx.x) >> 5);
  if (wave >= M_TILES) return;
  int lane = threadIdx.x & 31;
  int mr = lane & 15;
  int kh = lane >> 4;
  const float* arow = h + (size_t)(wave * 16 + mr) * HIDDEN;
  const float* b0r = dw + (size_t)(mr +  0) * HIDDEN;
  const float* b1r = dw + (size_t)(mr + 16) * HIDDEN;
  int n2 = mr + 32;                       // tile 2 columns 32..47; valid only < 40
  const float* b2r = dw + (size_t)(n2 < NCLASS ? n2 : 0) * HIDDEN;
  float b2scale = (n2 < NCLASS) ? 1.0f : 0.0f;
  v8f c0 = {}, c1 = {}, c2 = {};
  for (int k = 0; k < HIDDEN; k += 4) {
    int ko = k + 2 * kh;
    v2f a  = *(const v2f*)(arow + ko);
    v2f b0 = *(const v2f*)(b0r + ko);
    v2f b1 = *(const v2f*)(b1r + ko);
    v2f b2 = *(const v2f*)(b2r + ko);
    b2 *= b2scale;                        // zero-pad columns >= 40 without divergence
    c0 = __builtin_amdgcn_wmma_f32_16x16x4_f32(false, a, false, b0, (short)0, c0, false, false);
    c1 = __builtin_amdgcn_wmma_f32_16x16x4_f32(false, a, false, b1, (short)0, c1, false, false);
    c2 = __builtin_amdgcn_wmma_f32_16x16x4_f32(false, a, false, b2, (short)0, c2, false, false);
  }
  int mbase = wave * 16 + kh * 8;
  for (int r = 0; r < 8; ++r) {
    float* orow = out + (size_t)(mbase + r) * NCLASS;
    orow[mr]      = c0[r];
    orow[mr + 16] = c1[r];
    if (n2 < NCLASS) orow[n2] = c2[r];
  }
}

// ---------------- launcher ----------------

extern "C" void kernel_launch(void* const* d_in, const int* in_sizes, int n_in,
                              void* d_out, int out_size, void* d_ws, size_t ws_size,
                              hipStream_t stream) {
  (void)in_sizes; (void)n_in; (void)out_size; (void)ws_size;
  const float* x     = (const float*)d_in[0];
  const int*   ei    = (const int*)  d_in[1];
  const float* enc_w = (const float*)d_in[2];
  const float* dec_w = (const float*)d_in[3];
  const float* ext_w = (const float*)d_in[4];
  const float* beta  = (const float*)d_in[5];
  const float* pw    = (const float*)d_in[6];
  const int* row = ei;
  const int* col = ei + N_EDGES;

  float* ws   = (float*)d_ws;
  float* deg  = ws;                                  // 50000 (deg -> deg_inv in place)
  float* coef = deg  + N_NODES;                      // 1.6M
  float* W    = coef + N_EDGES;                      // 4096
  float* h    = W    + HIDDEN * HIDDEN;              // 3.2M
  float* x0   = h    + (size_t)N_NODES * HIDDEN;     // 3.2M
  float* outp = x0   + (size_t)N_NODES * HIDDEN;     // 3.2M
  float* agg  = outp + (size_t)N_NODES * HIDDEN;     // 3.2M  (~58 MB total)

  hipMemsetAsync(deg, 0, N_NODES * sizeof(float), stream);
  deg_count_kernel<<<(N_EDGES + 255) / 256, 256, 0, stream>>>(col, deg);
  deg_inv_kernel<<<(N_NODES + 255) / 256, 256, 0, stream>>>(deg);
  coef_kernel<<<(N_EDGES + 255) / 256, 256, 0, stream>>>(row, col, deg, coef);
  weff_kernel<<<1, HIDDEN, 0, stream>>>(pw, W);

  const int gemm_blocks = (M_TILES + 3) / 4;         // 4 waves / 128-thread block
  enc_gemm_kernel<<<gemm_blocks, 128, 0, stream>>>(x, enc_w, h, x0);

  for (int l = 0; l < NUM_LAYERS; ++l) {
    hw_gemm_kernel<<<gemm_blocks, 128, 0, stream>>>(h, W, outp);
    hipMemsetAsync(agg, 0, (size_t)N_NODES * HIDDEN * sizeof(float), stream);
    edge_agg_kernel<<<N_EDGES / 8, 256, 0, stream>>>(row, col, coef, outp, agg);
    update_kernel<<<(N_NODES * HIDDEN) / 256, 256, 0, stream>>>(h, agg, x0, ext_w, beta);
  }

  dec_gemm_kernel<<<gemm_blocks, 128, 0, stream>>>(h, dec_w, (float*)d_out);
}